// DGCNN_89335319757075
// MI455X (gfx1250) — compile-verified
//
#include <hip/hip_runtime.h>
#include <hip/hip_bf16.h>

// ---------------- problem constants ----------------
#define BATCH 8
#define CIN   4
#define NPTS  2048
#define KNN   20
#define OUTC  40

typedef float v2f __attribute__((ext_vector_type(2)));
typedef float v8f __attribute__((ext_vector_type(8)));

// ===================================================================
// Kernel 1: KNN (top-20 by -||xi-xj||^2) -> idx (B,N,K)
// One thread per row n; x[b] staged in LDS (broadcast reads across wave).
// ===================================================================
__global__ void knn_kernel(const float* __restrict__ x, int* __restrict__ idx) {
    __shared__ float xs[CIN][NPTS];                      // 32 KB
    const int b  = blockIdx.x >> 5;                      // N/64 = 32 blocks per batch
    const int n0 = (blockIdx.x & 31) * 64;
    for (int t = threadIdx.x; t < CIN * NPTS; t += 64) {
        int c = t >> 11, m = t & (NPTS - 1);
        xs[c][m] = x[(b * CIN + c) * NPTS + m];
    }
    __syncthreads();

    const int n = n0 + threadIdx.x;
    const float x0 = xs[0][n], x1 = xs[1][n], x2 = xs[2][n], x3 = xs[3][n];

    float best[KNN];
    int   bidx[KNN];
#pragma unroll
    for (int j = 0; j < KNN; ++j) { best[j] = -3.4e38f; bidx[j] = 0; }

    for (int m = 0; m < NPTS; ++m) {
        float d0 = x0 - xs[0][m], d1 = x1 - xs[1][m];
        float d2 = x2 - xs[2][m], d3 = x3 - xs[3][m];
        float pd = -(d0 * d0 + d1 * d1 + d2 * d2 + d3 * d3);
        if (pd > best[KNN - 1]) {
            int j = KNN - 1;
            while (j > 0 && best[j - 1] < pd) {      // strict: ties keep lower index first
                best[j] = best[j - 1]; bidx[j] = bidx[j - 1]; --j;
            }
            best[j] = pd; bidx[j] = m;
        }
    }
    int* op = idx + (b * NPTS + n) * KNN;
#pragma unroll
    for (int j = 0; j < KNN; ++j) op[j] = bidx[j];
}

// ===================================================================
// WMMA f32 16x16x4 tile GEMM over LDS/global operands.
//   dst[m][ch] = relu(scale[ch]*(sum_k src[m][k]*w[ch][k]) + shift[ch])
// A layout (ISA 7.12.2): vgpr0 = K = 2*half, vgpr1 = K = 2*half+1, M = lane%16.
// B mirrors A (K rows striped by half-wave), N = lane%16.
// D: vgpr r -> M = mt*16 + r + 8*half, N = nt*16 + lane%16.
// ===================================================================
template <int SRC_PITCH, int DST_PITCH, int MTILES, int NTILES, int KCHUNKS, int NWAVES>
__device__ inline void wmma_conv(const float* __restrict__ src,
                                 const float* __restrict__ wsrc, int wPitch,
                                 const float* __restrict__ scale,
                                 const float* __restrict__ shift,
                                 float* __restrict__ dst) {
    const int lane = threadIdx.x & 31;
    const int wave = threadIdx.x >> 5;
    const int half = lane >> 4;
    const int l16  = lane & 15;
    constexpr int PER = (MTILES * NTILES) / NWAVES;      // uniform => EXEC all-1s at WMMA
    for (int tt = 0; tt < PER; ++tt) {
        const int tile = wave * PER + tt;
        const int mt = tile / NTILES, nt = tile % NTILES;
        const float* arow = src + (mt * 16 + l16) * SRC_PITCH;
        const float* brow = wsrc + (nt * 16 + l16) * wPitch;
        v8f acc = {};
#pragma unroll
        for (int kc = 0; kc < KCHUNKS; ++kc) {
            const int k0 = kc * 4 + half * 2;
            v2f a  = { arow[k0], arow[k0 + 1] };
            v2f bf = { brow[k0], brow[k0 + 1] };
            acc = __builtin_amdgcn_wmma_f32_16x16x4_f32(false, a, false, bf,
                                                        (short)0, acc, false, false);
        }
        const int ch = nt * 16 + l16;
        const float sc = scale[ch], sh = shift[ch];
#pragma unroll
        for (int r = 0; r < 8; ++r) {
            float v = acc[r] * sc + sh;
            dst[(mt * 16 + r + half * 8) * DST_PITCH + ch] = v > 0.f ? v : 0.f;
        }
    }
}

// ===================================================================
// Kernel 2: fused graph-feature + conv1 + conv2 + conv3 + max over K
// Block = 4 points x 20 nbrs = 80 positions = 5 M-tiles. 160 threads (5 waves).
// ===================================================================
#define TN 4
#define MPOS (TN * KNN)        // 80
#define AP 68                  // LDS pitch (64 + 4 -> conflict-free column reads)

__global__ void edgeconv_kernel(const float* __restrict__ x, const int* __restrict__ idx,
                                const float* __restrict__ w1, const float* __restrict__ s1, const float* __restrict__ b1,
                                const float* __restrict__ w2, const float* __restrict__ s2, const float* __restrict__ b2,
                                const float* __restrict__ w3, const float* __restrict__ s3, const float* __restrict__ b3,
                                float* __restrict__ h3max) {
    __shared__ float hA[MPOS][AP];   // 21.8 KB
    __shared__ float hB[MPOS][AP];   // 21.8 KB
    __shared__ float wS[64][AP];     // 17.4 KB   (total ~61 KB)

    const int b  = blockIdx.x / (NPTS / TN);
    const int n0 = (blockIdx.x % (NPTS / TN)) * TN;
    const int tid = threadIdx.x;     // 160 threads

    // ---- phase 1: gather edge feature (8ch) + conv1 + bn1 + relu -> hA ----
    for (int t = tid; t < MPOS * 64; t += 160) {
        const int p = t >> 6, o = t & 63;
        const int tn = p / KNN, k = p % KNN;
        const int n  = n0 + tn;
        const int nb = idx[(b * NPTS + n) * KNN + k];
        float acc = 0.f;
#pragma unroll
        for (int c = 0; c < CIN; ++c) {
            acc = fmaf(w1[o * 8 + c],        x[(b * CIN + c) * NPTS + n],  acc);
            acc = fmaf(w1[o * 8 + CIN + c],  x[(b * CIN + c) * NPTS + nb], acc);
        }
        acc = acc * s1[o] + b1[o];
        hA[p][o] = acc > 0.f ? acc : 0.f;
    }
    for (int t = tid; t < 64 * 64; t += 160) wS[t >> 6][t & 63] = w2[t];
    __syncthreads();

    // ---- phase 2: conv2 (64x64) via WMMA ----
    wmma_conv<AP, AP, 5, 4, 16, 5>(&hA[0][0], &wS[0][0], AP, s2, b2, &hB[0][0]);
    __syncthreads();
    for (int t = tid; t < 64 * 64; t += 160) wS[t >> 6][t & 63] = w3[t];
    __syncthreads();

    // ---- phase 3: conv3 (64x64) via WMMA ----
    wmma_conv<AP, AP, 5, 4, 16, 5>(&hB[0][0], &wS[0][0], AP, s3, b3, &hA[0][0]);
    __syncthreads();

    // ---- phase 4: max over K neighbors -> h3max (B,64,N) ----
    for (int t = tid; t < TN * 64; t += 160) {
        const int tn = t >> 6, o = t & 63;
        float m = hA[tn * KNN][o];
#pragma unroll
        for (int k = 1; k < KNN; ++k) m = fmaxf(m, hA[tn * KNN + k][o]);
        h3max[(b * 64 + o) * NPTS + n0 + tn] = m;
    }
}

// ===================================================================
// Kernel 3: conv4 (128x64) + conv5 (1024x128) + global max pool -> g (B,1024)
// Block = 64 positions (4 M-tiles), 256 threads (8 waves). w5 streamed from L2.
// ===================================================================
#define HP 132                 // 128 + 4 pitch

__global__ void conv45_kernel(const float* __restrict__ h3max,
                              const float* __restrict__ w4, const float* __restrict__ s4, const float* __restrict__ b4,
                              const float* __restrict__ w5, const float* __restrict__ s5, const float* __restrict__ b5,
                              float* __restrict__ g) {
    __shared__ float a4[64][AP];     // 17.4 KB
    __shared__ float h4[64][HP];     // 33.8 KB
    __shared__ int   gmaxS[1024];    //  4.0 KB  (total ~55 KB)

    const int b  = blockIdx.x >> 5;                  // N/64 = 32 blocks per batch
    const int n0 = (blockIdx.x & 31) * 64;
    const int tid = threadIdx.x;

    for (int t = tid; t < 64 * 64; t += 256) {       // coalesced over n
        const int c = t >> 6, p = t & 63;
        a4[p][c] = h3max[(b * 64 + c) * NPTS + n0 + p];
    }
    for (int t = tid; t < 1024; t += 256) gmaxS[t] = 0;
    __syncthreads();

    // conv4: M=4 tiles, N=8 tiles (128ch), K=64 (16 chunks); weights from global
    wmma_conv<AP, HP, 4, 8, 16, 8>(&a4[0][0], w4, 64, s4, b4, &h4[0][0]);
    __syncthreads();

    // conv5: M=4 tiles, N=64 tiles (1024ch), K=128 (32 chunks); fold ReLU + max
    {
        const int lane = tid & 31, wave = tid >> 5;
        const int half = lane >> 4, l16 = lane & 15;
        for (int tt = 0; tt < 32; ++tt) {            // 256 tiles / 8 waves
            const int tile = wave * 32 + tt;
            const int mt = tile >> 6, nt = tile & 63;
            const float* arow = &h4[mt * 16 + l16][0];
            const float* brow = w5 + (nt * 16 + l16) * 128;
            v8f acc = {};
#pragma unroll
            for (int kc = 0; kc < 32; ++kc) {
                const int k0 = kc * 4 + half * 2;
                v2f a  = { arow[k0], arow[k0 + 1] };
                v2f bf = { brow[k0], brow[k0 + 1] };
                acc = __builtin_amdgcn_wmma_f32_16x16x4_f32(false, a, false, bf,
                                                            (short)0, acc, false, false);
            }
            const int ch = nt * 16 + l16;
            const float sc = s5[ch], sh = b5[ch];
            float mymax = 0.f;                       // relu floor; matches max(relu(.))
#pragma unroll
            for (int r = 0; r < 8; ++r) {
                float v = acc[r] * sc + sh;
                mymax = fmaxf(mymax, v > 0.f ? v : 0.f);
            }
            mymax = fmaxf(mymax, __shfl_xor(mymax, 16, 32));   // merge M halves
            if (half == 0) atomicMax(&gmaxS[ch], __float_as_int(mymax)); // >=0 => bit order ok
        }
    }
    __syncthreads();
    for (int t = tid; t < 1024; t += 256)
        atomicMax((int*)(g + b * 1024 + t), gmaxS[t]);
}

// ===================================================================
// Kernel 4: generic linear (+optional bias, +optional folded-BN, +optional relu)
// ===================================================================
__global__ void zero_kernel(float* __restrict__ p, int n) {
    int t = blockIdx.x * blockDim.x + threadIdx.x;
    if (t < n) p[t] = 0.f;
}

__global__ void mlp_kernel(const float* __restrict__ in, const float* __restrict__ w,
                           const float* __restrict__ bias, const float* __restrict__ scale,
                           const float* __restrict__ shift, float* __restrict__ out,
                           int IN, int ON, int doRelu) {
    const int t = blockIdx.x * blockDim.x + threadIdx.x;
    if (t >= BATCH * ON) return;
    const int b = t / ON, o = t % ON;
    const float* ip = in + b * IN;
    const float* wp = w  + o * IN;
    float acc = bias ? bias[o] : 0.f;
    for (int i = 0; i < IN; ++i) acc = fmaf(ip[i], wp[i], acc);
    if (scale) acc = acc * scale[o] + shift[o];
    out[t] = doRelu ? fmaxf(acc, 0.f) : acc;
}

// ===================================================================
// host launcher
// ===================================================================
extern "C" void kernel_launch(void* const* d_in, const int* in_sizes, int n_in,
                              void* d_out, int out_size, void* d_ws, size_t ws_size,
                              hipStream_t stream) {
    const float* x   = (const float*)d_in[0];
    const float* w1  = (const float*)d_in[1];
    const float* s1  = (const float*)d_in[2];
    const float* b1  = (const float*)d_in[3];
    const float* w2  = (const float*)d_in[4];
    const float* s2  = (const float*)d_in[5];
    const float* b2  = (const float*)d_in[6];
    const float* w3  = (const float*)d_in[7];
    const float* s3  = (const float*)d_in[8];
    const float* b3  = (const float*)d_in[9];
    const float* w4  = (const float*)d_in[10];
    const float* s4  = (const float*)d_in[11];
    const float* b4  = (const float*)d_in[12];
    const float* w5  = (const float*)d_in[13];
    const float* s5  = (const float*)d_in[14];
    const float* b5  = (const float*)d_in[15];
    const float* l1w = (const float*)d_in[16];
    const float* s6  = (const float*)d_in[17];
    const float* b6  = (const float*)d_in[18];
    const float* l2w = (const float*)d_in[19];
    const float* l2b = (const float*)d_in[20];
    const float* s7  = (const float*)d_in[21];
    const float* b7  = (const float*)d_in[22];
    const float* l3w = (const float*)d_in[23];
    const float* l3b = (const float*)d_in[24];

    // workspace carve-out
    char* ws = (char*)d_ws;
    const size_t IDX_BYTES = (size_t)BATCH * NPTS * KNN * sizeof(int);       // 1.31 MB
    const size_t H3_BYTES  = (size_t)BATCH * 64 * NPTS * sizeof(float);      // 4.19 MB
    const size_t G_BYTES   = (size_t)BATCH * 1024 * sizeof(float);
    const size_t G1_BYTES  = (size_t)BATCH * 512 * sizeof(float);
    int*   idx   = (int*)ws;
    float* h3max = (float*)(ws + IDX_BYTES);
    float* g     = (float*)(ws + IDX_BYTES + H3_BYTES);
    float* g1    = (float*)(ws + IDX_BYTES + H3_BYTES + G_BYTES);
    float* g2    = (float*)(ws + IDX_BYTES + H3_BYTES + G_BYTES + G1_BYTES);

    // 1) knn indices
    knn_kernel<<<BATCH * (NPTS / 64), 64, 0, stream>>>(x, idx);
    // zero global-max accumulator (relu outputs are >= 0)
    zero_kernel<<<(BATCH * 1024 + 255) / 256, 256, 0, stream>>>(g, BATCH * 1024);
    // 2) graph-feature + conv1..3 + max over K
    edgeconv_kernel<<<BATCH * (NPTS / TN), 160, 0, stream>>>(
        x, idx, w1, s1, b1, w2, s2, b2, w3, s3, b3, h3max);
    // 3) conv4 + conv5 + global max pool
    conv45_kernel<<<BATCH * (NPTS / 64), 256, 0, stream>>>(
        h3max, w4, s4, b4, w5, s5, b5, g);
    // 4) MLP head
    mlp_kernel<<<(BATCH * 512 + 255) / 256, 256, 0, stream>>>(g,  l1w, nullptr, s6, b6, g1, 1024, 512, 1);
    mlp_kernel<<<(BATCH * 256 + 255) / 256, 256, 0, stream>>>(g1, l2w, l2b,     s7, b7, g2, 512, 256, 1);
    mlp_kernel<<<(BATCH * OUTC + 255) / 256, 256, 0, stream>>>(g2, l3w, l3b, nullptr, nullptr,
                                                               (float*)d_out, 256, OUTC, 0);
}